// TropicalConv2D_88338887344627
// MI455X (gfx1250) — compile-verified
//
#include <hip/hip_runtime.h>
#include <cstdint>

#ifndef __has_builtin
#define __has_builtin(x) 0
#endif

#define TILE 32
#define HALO 2
#define LW (TILE + 2 * HALO)      // 36
#define LDS_ELEMS (LW * LW)       // 1296
#define NTHREADS 256

// ---- CDNA5 async global->LDS copy (ASYNCcnt path) -------------------------

__device__ __forceinline__ void async_g2l_b32(void* lds_dst, const void* gsrc) {
#if __has_builtin(__builtin_amdgcn_global_load_async_to_lds_b32)
  __builtin_amdgcn_global_load_async_to_lds_b32(
      (__attribute__((address_space(1))) int*)gsrc,
      (__attribute__((address_space(3))) int*)lds_dst,
      /*imm offset=*/0, /*cpol=*/0);
#else
  unsigned loff = (unsigned)(uintptr_t)(__attribute__((address_space(3))) int*)lds_dst;
  asm volatile("global_load_async_to_lds_b32 %0, %1, off"
               :: "v"(loff), "v"(gsrc) : "memory");
#endif
}

__device__ __forceinline__ void wait_asynccnt0() {
#if __has_builtin(__builtin_amdgcn_s_wait_asynccnt)
  __builtin_amdgcn_s_wait_asynccnt(0);
#else
  asm volatile("s_wait_asynccnt 0" ::: "memory");
#endif
}

// ---- Tropical (max-plus) depthwise 5x5 conv, stride 1, pad 2 --------------

__global__ __launch_bounds__(NTHREADS)
void tropical_conv2d_kernel(const float* __restrict__ x,
                            const float* __restrict__ kern,
                            float* __restrict__ out) {
  constexpr int C = 32, H = 224, W = 224;
  constexpr int TPR = W / TILE;   // 7 tiles per row

  __shared__ float tile[LDS_ELEMS];
  __shared__ float wk[25];

  const int tid = threadIdx.x;
  const int tIdx = blockIdx.x;          // 0..48
  const int c = blockIdx.y;
  const int b = blockIdx.z;
  const int tr = tIdx / TPR;
  const int tc = tIdx - tr * TPR;
  const int r0 = tr * TILE;
  const int c0 = tc * TILE;

  const float* xplane = x + (size_t)(b * C + c) * (H * W);
  float*       oplane = out + (size_t)(b * C + c) * (H * W);
  const float* kplane = kern + c * 25;

  const float NEG = -__builtin_inff();

  // Phase 0: pre-fill halo tile with -inf (implements -inf padding).
  for (int e = tid; e < LDS_ELEMS; e += NTHREADS) tile[e] = NEG;
  __syncthreads();

  // Phase 1: async-stage valid input elements + per-channel weights into LDS.
  for (int e = tid; e < LDS_ELEMS; e += NTHREADS) {
    const int lr = e / LW;
    const int lc = e - lr * LW;
    const int gr = r0 - HALO + lr;
    const int gc = c0 - HALO + lc;
    if ((unsigned)gr < (unsigned)H && (unsigned)gc < (unsigned)W) {
      async_g2l_b32(&tile[e], xplane + (size_t)gr * W + gc);
    }
  }
  if (tid < 25) {
    async_g2l_b32(&wk[tid], kplane + tid);
  }
  wait_asynccnt0();
  __syncthreads();

  // Phase 2: each thread computes a 4-row x 1-col output strip.
  const int tx = tid & (TILE - 1);     // 0..31 -> conflict-free LDS rows
  const int ty = tid >> 5;             // 0..7  (wave id)
  const int y0 = ty * 4;

  // Load the 8-row x 5-col LDS sub-window once (40 ds_load_b32).
  float colv[5][8];
#pragma unroll
  for (int j = 0; j < 5; ++j) {
#pragma unroll
    for (int rr = 0; rr < 8; ++rr) {
      colv[j][rr] = tile[(y0 + rr) * LW + (tx + j)];
    }
  }

  float acc0 = NEG, acc1 = NEG, acc2 = NEG, acc3 = NEG;
#pragma unroll
  for (int i = 0; i < 5; ++i) {
#pragma unroll
    for (int j = 0; j < 5; ++j) {
      const float w = wk[(4 - i) * 5 + (4 - j)];   // flipped kernel
      acc0 = fmaxf(acc0, colv[j][i + 0] + w);
      acc1 = fmaxf(acc1, colv[j][i + 1] + w);
      acc2 = fmaxf(acc2, colv[j][i + 2] + w);
      acc3 = fmaxf(acc3, colv[j][i + 3] + w);
    }
  }

  float* op = oplane + (size_t)(r0 + y0) * W + (c0 + tx);
  __builtin_nontemporal_store(acc0, op);
  __builtin_nontemporal_store(acc1, op + (size_t)W);
  __builtin_nontemporal_store(acc2, op + (size_t)2 * W);
  __builtin_nontemporal_store(acc3, op + (size_t)3 * W);
}

// ---- Launch ---------------------------------------------------------------

extern "C" void kernel_launch(void* const* d_in, const int* in_sizes, int n_in,
                              void* d_out, int out_size, void* d_ws, size_t ws_size,
                              hipStream_t stream) {
  (void)in_sizes; (void)n_in; (void)out_size; (void)d_ws; (void)ws_size;
  const float* x    = (const float*)d_in[0];
  const float* kern = (const float*)d_in[1];
  float*       out  = (float*)d_out;

  constexpr int B = 8, C = 32, H = 224, W = 224;
  dim3 grid((H / TILE) * (W / TILE), C, B);   // (49, 32, 8)
  dim3 block(NTHREADS);
  hipLaunchKernelGGL(tropical_conv2d_kernel, grid, block, 0, stream, x, kern, out);
}